// GCNModel_12584254177713
// MI455X (gfx1250) — compile-verified
//
#include <hip/hip_runtime.h>
#include <hip/hip_bf16.h>
#include <stdint.h>

typedef __attribute__((ext_vector_type(2))) float v2f;
typedef __attribute__((ext_vector_type(4))) float v4f;
typedef __attribute__((ext_vector_type(8))) float v8f;

#define D_IN   512
#define D_HID  16
#define N_OUT  40
#define KCHUNK 32
#define ASTRIDE 36            // padded LDS row stride: bank-conflict-free, 16B-aligned
#define NCHUNK (D_IN / KCHUNK)

// ---------------- zero scratch ----------------
__global__ void zero_f32(float* p, long long n) {
    long long i = (long long)blockIdx.x * blockDim.x + threadIdx.x;
    if (i < n) p[i] = 0.0f;
}

// ---------------- degree: deg[dst] += w ; deg[i] += 1 (self loops) ----------------
__global__ void degree_kernel(const long long* __restrict__ dst,
                              const float* __restrict__ w,
                              float* __restrict__ deg, long long nE, int nN) {
    long long i = (long long)blockIdx.x * blockDim.x + threadIdx.x;
    if (i < nE) {
        atomicAdd(&deg[dst[i]], w[i]);
    } else if (i < nE + nN) {
        atomicAdd(&deg[(int)(i - nE)], 1.0f);
    }
}

// ---------------- dinv = deg>0 ? rsqrt(deg) : 0 (in place) ----------------
__global__ void dinv_kernel(float* deg, int nN) {
    int i = blockIdx.x * blockDim.x + threadIdx.x;
    if (i < nN) {
        float d = deg[i];
        deg[i] = (d > 0.0f) ? rsqrtf(d) : 0.0f;
    }
}

// ---------------- h1 = x @ W1 via V_WMMA_F32_16X16X4_F32 ----------------
// block = 128 threads (4 waves); wave w handles 16-row tile (blockIdx*4 + w).
// W1 (512x16 f32, 32KB) staged once in LDS. x tiles streamed with
// GLOBAL_LOAD_ASYNC_TO_LDS_B128, double-buffered, pipelined on ASYNCcnt.
__global__ void __launch_bounds__(128)
gemm1_wmma(const float* __restrict__ x, const float* __restrict__ W1,
           float* __restrict__ h1, int nRows) {
    __shared__ float sW[D_IN * D_HID];             // 32 KB
    __shared__ float sA[4][2][16 * ASTRIDE];       // 4 waves x 2 bufs x 2304 B = 18 KB

    const int tid = threadIdx.x;
    for (int i = tid; i < D_IN * D_HID; i += 128) sW[i] = W1[i];
    __syncthreads();

    const int wave = tid >> 5;
    const int lane = tid & 31;
    const int hl   = lane & 15;    // half-lane index
    const int hi   = lane >> 4;    // 0 / 1 half select
    const int rowBase = (blockIdx.x * 4 + wave) * 16;
    if (rowBase >= nRows) return;

    // async staging: 4 b128 transfers cover a 16x32 fp32 chunk
    // (each instr: 32 lanes x 16B = 4 rows x 128B)
    const int srow = lane >> 3;         // row-in-group 0..3
    const int scol = (lane & 7) * 4;    // float column 0,4,...,28
    const float* xbase = x + (size_t)rowBase * D_IN;   // loop-invariant SGPR base
    unsigned ldsA0 = (unsigned)(uintptr_t)(&sA[wave][0][0]);  // flat addr[31:0] == LDS offset
    unsigned ldsA1 = (unsigned)(uintptr_t)(&sA[wave][1][0]);

    v8f acc = {0.f, 0.f, 0.f, 0.f, 0.f, 0.f, 0.f, 0.f};

    auto stage = [&](int kc, unsigned ldsBase) {
        #pragma unroll
        for (int g = 0; g < 4; ++g) {
            const int row  = g * 4 + srow;
            unsigned goff  = (unsigned)((row * D_IN + kc * KCHUNK + scol) * 4);
            unsigned loff  = ldsBase + (unsigned)((row * ASTRIDE + scol) * 4);
            asm volatile("global_load_async_to_lds_b128 %0, %1, %2 offset:0"
                         :: "v"(loff), "v"(goff), "s"(xbase) : "memory");
        }
    };
    auto compute = [&](int kc, const float* sAw) {
        #pragma unroll
        for (int kk = 0; kk < KCHUNK; kk += 4) {
            const int K = kc * KCHUNK + kk;
            // A frag: lanes 0-15 -> {A[m][K],A[m][K+1]}, lanes 16-31 -> {A[m][K+2],A[m][K+3]}
            v2f a = *(const v2f*)(&sAw[hl * ASTRIDE + kk + 2 * hi]);
            // B frag: v0 = B[K+2*hi][n], v1 = B[K+1+2*hi][n]
            v2f b;
            b.x = sW[(K + 2 * hi)     * D_HID + hl];
            b.y = sW[(K + 2 * hi + 1) * D_HID + hl];
            acc = __builtin_amdgcn_wmma_f32_16x16x4_f32(
                false, a, false, b, (short)0, acc, false, false);
        }
    };

    stage(0, ldsA0);
    for (int kc = 0; kc < NCHUNK; ++kc) {
        const int buf = kc & 1;
        if (kc + 1 < NCHUNK) {
            // WAR: fragment reads from the buffer we are about to refill are done
            asm volatile("s_wait_dscnt 0x0" ::: "memory");
            stage(kc + 1, buf ? ldsA0 : ldsA1);
            // RAW: current chunk's 4 async transfers complete (in-order), next 4 in flight
            asm volatile("s_wait_asynccnt 0x4" ::: "memory");
        } else {
            asm volatile("s_wait_asynccnt 0x0" ::: "memory");
        }
        compute(kc, &sA[wave][buf][0]);
    }

    // C layout: VGPR j holds row (j + 8*hi), col hl
    #pragma unroll
    for (int j = 0; j < 8; ++j) {
        int m = rowBase + j + 8 * hi;
        h1[(size_t)m * D_HID + hl] = acc[j];
    }
}

// ---------------- edge aggregation: out[dst] += (feat[src] (+bias)) * norm ----------------
// one thread per (edge, feature); edges >= nE are self-loops (w=1, s=d=e-nE)
__global__ void aggregate(const long long* __restrict__ src,
                          const long long* __restrict__ dst,
                          const float* __restrict__ w,
                          const float* __restrict__ dinv,
                          const float* __restrict__ feat,
                          const float* __restrict__ bias,  // may be null
                          float* __restrict__ out,
                          long long nE, int nN) {
    long long t = (long long)blockIdx.x * blockDim.x + threadIdx.x;
    long long e = t >> 4;
    int j = (int)(t & 15);
    if (e >= nE + nN) return;
    long long s, d;
    float ww;
    if (e < nE) { s = src[e]; d = dst[e]; ww = w[e]; }
    else        { s = d = e - nE;          ww = 1.0f; }
    float nrm = dinv[s] * ww * dinv[d];
    float v = feat[s * D_HID + j];
    if (bias) v += bias[j];
    atomicAdd(&out[d * D_HID + j], v * nrm);
}

// ---------------- out = log_softmax(agg2 @ W2 + b2) ----------------
// one thread per node; W2/b2 accesses are wave-uniform -> scalar loads
__global__ void final_kernel(const float* __restrict__ agg2,
                             const float* __restrict__ W2,
                             const float* __restrict__ b2,
                             float* __restrict__ out, int nN) {
    int i = blockIdx.x * blockDim.x + threadIdx.x;
    if (i >= nN) return;
    float a[D_HID];
    const v4f* ap = (const v4f*)(agg2 + (size_t)i * D_HID);
    #pragma unroll
    for (int q = 0; q < 4; ++q) {
        v4f t = ap[q];
        a[4*q] = t.x; a[4*q+1] = t.y; a[4*q+2] = t.z; a[4*q+3] = t.w;
    }
    float o[N_OUT];
    float mx = -3.402823466e38f;
    #pragma unroll
    for (int c = 0; c < N_OUT; ++c) {
        float s = b2[c];
        #pragma unroll
        for (int k = 0; k < D_HID; ++k) s = fmaf(a[k], W2[k * N_OUT + c], s);
        o[c] = s;
        mx = fmaxf(mx, s);
    }
    float se = 0.0f;
    #pragma unroll
    for (int c = 0; c < N_OUT; ++c) se += __expf(o[c] - mx);
    float lse = mx + __logf(se);
    #pragma unroll
    for (int c = 0; c < N_OUT; ++c) out[(size_t)i * N_OUT + c] = o[c] - lse;
}

extern "C" void kernel_launch(void* const* d_in, const int* in_sizes, int n_in,
                              void* d_out, int out_size, void* d_ws, size_t ws_size,
                              hipStream_t stream) {
    const float*     x  = (const float*)d_in[0];
    const long long* ei = (const long long*)d_in[1];   // int64 [2, E]
    const float*     ea = (const float*)d_in[2];
    const float*     W1 = (const float*)d_in[3];
    const float*     b1 = (const float*)d_in[4];
    const float*     W2 = (const float*)d_in[5];
    const float*     b2 = (const float*)d_in[6];
    float* out = (float*)d_out;

    const int       N = in_sizes[0] / D_IN;
    const long long E = (long long)in_sizes[1] / 2;
    const long long* src = ei;
    const long long* dst = ei + E;

    // workspace layout (fp32): deg/dinv [N] | h1 [16N] | agg1 [16N] | agg2 [16N]
    float* ws   = (float*)d_ws;
    float* deg  = ws;
    float* h1   = ws + N;
    float* agg1 = ws + (size_t)17 * N;
    float* agg2 = ws + (size_t)33 * N;

    const long long totalZero = (long long)49 * N;
    zero_f32<<<(int)((totalZero + 255) / 256), 256, 0, stream>>>(ws, totalZero);

    const long long degT = E + N;
    degree_kernel<<<(int)((degT + 255) / 256), 256, 0, stream>>>(dst, ea, deg, E, N);
    dinv_kernel<<<(N + 255) / 256, 256, 0, stream>>>(deg, N);

    const int nTiles  = (N + 15) / 16;       // N=100000 -> exactly 6250 tiles
    const int nBlocks = (nTiles + 3) / 4;
    gemm1_wmma<<<nBlocks, 128, 0, stream>>>(x, W1, h1, N);

    const long long aggT = (E + N) * D_HID;
    const int aggB = (int)((aggT + 255) / 256);
    // layer 1: agg1 = A_norm @ h1
    aggregate<<<aggB, 256, 0, stream>>>(src, dst, ea, deg, h1, nullptr, agg1, E, N);
    // layer 2 (W2 hoisted past aggregation): agg2 = A_norm @ (agg1 + b1)
    aggregate<<<aggB, 256, 0, stream>>>(src, dst, ea, deg, agg1, b1, agg2, E, N);

    final_kernel<<<(N + 255) / 256, 256, 0, stream>>>(agg2, W2, b2, out, N);
}